// RelationalLayer_86706799772159
// MI455X (gfx1250) — compile-verified
//
#include <hip/hip_runtime.h>
#include <hip/hip_bf16.h>

typedef __attribute__((ext_vector_type(2))) float v2f;
typedef __attribute__((ext_vector_type(8))) float v8f;

// ---------------------------------------------------------------------------
// NT GEMM via V_WMMA_F32_16X16X4_F32:  C[M,N] = A[M,K] * B[N,K]^T  (all f32)
// Block = 256 threads = 8 waves (wave32), arranged 2x4; each wave owns a
// 64x64 C tile (4x4 WMMA accumulators). Grid.x = M/128; N is fixed at 256.
// Fragment layouts per CDNA5 ISA 7.12.2:
//   A 16x4 f32 : lane m = lane%16, VGPR0/1 hold K = 2*(lane/16) + {0,1}
//   B 4x16 f32 : lane n = lane%16, VGPR0/1 hold K = 2*(lane/16) + {0,1}
//   C 16x16 f32: VGPR v, lane -> row = v + 8*(lane/16), col = lane%16
// ---------------------------------------------------------------------------
__global__ __launch_bounds__(256) void gemm_nt_wmma_f32(
    const float* __restrict__ A, const float* __restrict__ B,
    float* __restrict__ C, int K, int lda, int ldb, int ldc)
{
    const int lane = threadIdx.x & 31;
    const int wave = threadIdx.x >> 5;
    const int wrow = wave >> 2;          // 0..1
    const int wcol = wave & 3;           // 0..3
    const int m0   = blockIdx.x * 128 + wrow * 64;
    const int n0   = wcol * 64;
    const int lm   = lane & 15;          // row (A) / col (B,C) within 16
    const int lk   = (lane >> 4) << 1;   // K sub-offset: 0 or 2
    const int g    = lane >> 4;          // lane half

    v8f acc[4][4];
#pragma unroll
    for (int i = 0; i < 4; ++i)
#pragma unroll
        for (int j = 0; j < 4; ++j)
            acc[i][j] = (v8f){0.f, 0.f, 0.f, 0.f, 0.f, 0.f, 0.f, 0.f};

    const float* aBase = A + (size_t)(m0 + lm) * lda + lk;
    const float* bBase = B + (size_t)(n0 + lm) * ldb + lk;

    for (int k = 0; k < K; k += 4) {
        v2f af[4], bf[4];
#pragma unroll
        for (int i = 0; i < 4; ++i) {
            af[i] = *(const v2f*)(aBase + (size_t)(i * 16) * lda + k);
            bf[i] = *(const v2f*)(bBase + (size_t)(i * 16) * ldb + k);
        }
#pragma unroll
        for (int i = 0; i < 4; ++i)
#pragma unroll
            for (int j = 0; j < 4; ++j)
                acc[i][j] = __builtin_amdgcn_wmma_f32_16x16x4_f32(
                    /*neg_a=*/false, af[i], /*neg_b=*/false, bf[j],
                    /*c_mod=*/(short)0, acc[i][j],
                    /*reuse_a=*/false, /*reuse_b=*/false);
    }

#pragma unroll
    for (int i = 0; i < 4; ++i) {
#pragma unroll
        for (int j = 0; j < 4; ++j) {
            float* cp = C + (size_t)(m0 + i * 16 + 8 * g) * ldc + n0 + j * 16 + lm;
#pragma unroll
            for (int v = 0; v < 8; ++v)
                cp[(size_t)v * ldc] = acc[i][j][v];
        }
    }
}

// ---------------------------------------------------------------------------
// c[p] = b2[p] + sum_o W2[p,o] * b1[o]     (one 256-thread block, O = 256)
// ---------------------------------------------------------------------------
__global__ __launch_bounds__(256) void bias_fold_kernel(
    const float* __restrict__ W2, const float* __restrict__ b1,
    const float* __restrict__ b2, float* __restrict__ cvec)
{
    const int p = threadIdx.x;
    float acc = b2[p];
    const float* row = W2 + (size_t)p * 256;
    for (int o = 0; o < 256; ++o)
        acc = fmaf(row[o], b1[o], acc);
    cvec[p] = acc;
}

// ---------------------------------------------------------------------------
// out[b,i,p] = sum_j relu(Ab[b,i,p] + Bb[b,j,p] + c[p])
//            - relu(Ab[b,i,p] + Bb[b,i,p] + c[p])     (exact diagonal mask)
// Grid = B*N blocks (one per (b,i)), 256 threads (one per p).
// ---------------------------------------------------------------------------
__global__ __launch_bounds__(256) void pair_reduce_kernel(
    const float* __restrict__ Ab, const float* __restrict__ Bb,
    const float* __restrict__ cvec, float* __restrict__ out)
{
    const int bi = blockIdx.x;       // b*64 + i
    const int b  = bi >> 6;
    const int i  = bi & 63;
    const int p  = threadIdx.x;

    const float a = Ab[(size_t)bi * 256 + p] + cvec[p];
    const float* Brow = Bb + (size_t)b * 64 * 256 + p;

    float acc = 0.f;
#pragma unroll 8
    for (int j = 0; j < 64; ++j)
        acc += fmaxf(a + Brow[(size_t)j * 256], 0.f);

    acc -= fmaxf(a + Brow[(size_t)i * 256], 0.f);
    out[(size_t)bi * 256 + p] = acc;
}

extern "C" void kernel_launch(void* const* d_in, const int* in_sizes, int n_in,
                              void* d_out, int out_size, void* d_ws, size_t ws_size,
                              hipStream_t stream)
{
    (void)in_sizes; (void)n_in; (void)out_size; (void)ws_size;

    const float* x  = (const float*)d_in[0];  // (32,64,512)
    const float* W1 = (const float*)d_in[1];  // (256,1024) : Wa = [:, :512], Wb = [:, 512:]
    const float* b1 = (const float*)d_in[2];  // (256,)
    const float* W2 = (const float*)d_in[3];  // (256,256)
    const float* b2 = (const float*)d_in[4];  // (256,)
    float* out = (float*)d_out;               // (32,64,256)

    const int D = 512, O = 256;
    const int M = 32 * 64;                    // 2048 flattened (b,n)

    float* hi = (float*)d_ws;                 // (2048,256)
    float* hj = hi + (size_t)M * O;           // (2048,256)
    float* Ab = hj + (size_t)M * O;           // (2048,256)
    float* Bb = Ab + (size_t)M * O;           // (2048,256)
    float* cv = Bb + (size_t)M * O;           // (256,)

    dim3 block(256);
    dim3 grid(M / 128);                       // 16 blocks, each 128x256 of C

    // hi = x * Wa^T ; hj = x * Wb^T          (K = 512, ldb = 1024)
    gemm_nt_wmma_f32<<<grid, block, 0, stream>>>(x,  W1,     hi, D, D, 2 * D, O);
    gemm_nt_wmma_f32<<<grid, block, 0, stream>>>(x,  W1 + D, hj, D, D, 2 * D, O);
    // Ab = hi * W2^T ; Bb = hj * W2^T        (K = 256)
    gemm_nt_wmma_f32<<<grid, block, 0, stream>>>(hi, W2,     Ab, O, O, O,     O);
    gemm_nt_wmma_f32<<<grid, block, 0, stream>>>(hj, W2,     Bb, O, O, O,     O);
    // c = W2*b1 + b2
    bias_fold_kernel<<<1, 256, 0, stream>>>(W2, b1, b2, cv);
    // masked relu pairwise sum
    pair_reduce_kernel<<<M, 256, 0, stream>>>(Ab, Bb, cv, out);
}